// ListMLELoss_17540646437389
// MI455X (gfx1250) — compile-verified
//
#include <hip/hip_runtime.h>
#include <hip/hip_bf16.h>
#include <stdint.h>

#define TPB 256            // 8 wave32 waves per block
#define IPT 16             // items per thread
#define EPB (TPB * IPT)    // 4096 elements per block
#define BPT 16             // blocks per thread in single-block scans (NB == TPB*BPT)

typedef __attribute__((ext_vector_type(2))) float v2f;
typedef __attribute__((ext_vector_type(8))) float v8f;

// ---------- online-softmax pair combine: (m,s) with s = sum exp(x - m) ----------
__device__ __forceinline__ void lse_combine(float m1, float s1, float m2, float s2,
                                            float& mo, float& so) {
    float mm = fmaxf(m1, m2);
    float r;
    if (mm == -INFINITY) {
        r = 0.0f;
    } else {
        r = s1 * __expf(m1 - mm) + s2 * __expf(m2 - mm);
    }
    mo = mm;
    so = r;
}

// ---------------------------------------------------------------------------
// Prep: build sortable keys from labels (descending), copy scores as payload,
// per-block score sums (deterministic tree), NaN flag.
// ---------------------------------------------------------------------------
__global__ void k_prep(const float* __restrict__ labels, const float* __restrict__ scores,
                       uint32_t* __restrict__ keys, float* __restrict__ pay,
                       float* __restrict__ partialSum, int* __restrict__ nanFlag) {
    __shared__ float red[TPB];
    const int tid = threadIdx.x;
    const long long base = (long long)blockIdx.x * EPB + (long long)tid * IPT;
    const float4* l4 = (const float4*)(labels + base);
    const float4* s4 = (const float4*)(scores + base);
    uint4* k4 = (uint4*)(keys + base);
    float4* p4 = (float4*)(pay + base);

    float sum = 0.0f;
    bool anynan = false;
#pragma unroll
    for (int q = 0; q < 4; q++) {
        float4 lv = l4[q];
        float4 sv = s4[q];
        float lf[4] = {lv.x, lv.y, lv.z, lv.w};
        float sf[4] = {sv.x, sv.y, sv.z, sv.w};
        uint32_t kk[4];
#pragma unroll
        for (int j = 0; j < 4; j++) {
            uint32_t u = __float_as_uint(lf[j]);
            u ^= (u >> 31) ? 0xFFFFFFFFu : 0x80000000u;  // ascending-comparable
            kk[j] = ~u;                                  // invert -> labels descending
            sum += sf[j];
            anynan |= (sf[j] != sf[j]);
        }
        uint4 kv; kv.x = kk[0]; kv.y = kk[1]; kv.z = kk[2]; kv.w = kk[3];
        k4[q] = kv;
        p4[q] = sv;
    }
    if (anynan) atomicOr(nanFlag, 1);

    red[tid] = sum;
    __syncthreads();
    for (int off = TPB / 2; off > 0; off >>= 1) {
        if (tid < off) red[tid] += red[tid + off];
        __syncthreads();
    }
    if (tid == 0) partialSum[blockIdx.x] = red[0];
}

// ---------------------------------------------------------------------------
// Radix pass kernel 1: per-block 256-bin histogram (LDS atomics), bin-major out.
// ---------------------------------------------------------------------------
__global__ void k_hist(const uint32_t* __restrict__ keys, uint32_t* __restrict__ counts,
                       int shift, int numBlocks) {
    __shared__ uint32_t h[256];
    const int tid = threadIdx.x;
    h[tid] = 0;
    __syncthreads();
    const long long base = (long long)blockIdx.x * EPB + (long long)tid * IPT;
    const uint4* k4 = (const uint4*)(keys + base);
#pragma unroll
    for (int q = 0; q < 4; q++) {
        uint4 kk = k4[q];
        atomicAdd(&h[(kk.x >> shift) & 255u], 1u);
        atomicAdd(&h[(kk.y >> shift) & 255u], 1u);
        atomicAdd(&h[(kk.z >> shift) & 255u], 1u);
        atomicAdd(&h[(kk.w >> shift) & 255u], 1u);
    }
    __syncthreads();
    counts[(size_t)tid * numBlocks + blockIdx.x] = h[tid];
}

// ---------------------------------------------------------------------------
// Radix pass kernel 2: per-bin total (256 blocks, one bin each).
// ---------------------------------------------------------------------------
__global__ void k_bin_total(const uint32_t* __restrict__ counts, uint32_t* __restrict__ binTotal,
                            int numBlocks) {
    __shared__ uint32_t red[TPB];
    const int tid = threadIdx.x;
    const int b = blockIdx.x;
    uint32_t s = 0;
    for (int i = tid; i < numBlocks; i += TPB) s += counts[(size_t)b * numBlocks + i];
    red[tid] = s;
    __syncthreads();
    for (int off = TPB / 2; off > 0; off >>= 1) {
        if (tid < off) red[tid] += red[tid + off];
        __syncthreads();
    }
    if (tid == 0) binTotal[b] = red[0];
}

// ---------------------------------------------------------------------------
// Radix pass kernel 3: exclusive scan over the 256 bin totals (single block).
// ---------------------------------------------------------------------------
__global__ void k_bin_scan(const uint32_t* __restrict__ binTotal, uint32_t* __restrict__ binStart) {
    __shared__ uint32_t v[256];
    const int tid = threadIdx.x;
    const uint32_t own = binTotal[tid];
    v[tid] = own;
    __syncthreads();
    for (int off = 1; off < 256; off <<= 1) {
        uint32_t t = (tid >= off) ? v[tid - off] : 0u;
        __syncthreads();
        v[tid] += t;
        __syncthreads();
    }
    binStart[tid] = v[tid] - own;  // exclusive
}

// ---------------------------------------------------------------------------
// Radix pass kernel 4: per-bin exclusive scan across blocks -> global offsets
// (in place over counts). 256 blocks (one bin each), block-parallel scan.
// ---------------------------------------------------------------------------
__global__ void k_offsets(uint32_t* __restrict__ counts, const uint32_t* __restrict__ binStart,
                          int numBlocks) {
    __shared__ uint32_t tsum[TPB];
    const int tid = threadIdx.x;
    const int b = blockIdx.x;
    const size_t rb = (size_t)b * numBlocks;
    const int ipc = numBlocks / TPB;  // == BPT

    uint32_t c[BPT];
    uint32_t s = 0;
#pragma unroll
    for (int j = 0; j < BPT; j++) {
        c[j] = counts[rb + (size_t)tid * ipc + j];
        s += c[j];
    }
    tsum[tid] = s;
    __syncthreads();
    for (int off = 1; off < TPB; off <<= 1) {
        uint32_t t = (tid >= off) ? tsum[tid - off] : 0u;
        __syncthreads();
        tsum[tid] += t;
        __syncthreads();
    }
    uint32_t run = binStart[b] + tsum[tid] - s;  // exclusive base for this thread
#pragma unroll
    for (int j = 0; j < BPT; j++) {
        uint32_t cc = c[j];
        counts[rb + (size_t)tid * ipc + j] = run;
        run += cc;
    }
}

// ---------------------------------------------------------------------------
// Radix pass kernel 5: stable scatter. LDS multisplit: per-thread per-digit
// u16 counts (256x256 = 128KB, legal on CDNA5's 320KB WGP LDS), per-digit
// serial scan across threads, then ordered scatter. L2-resident traffic.
// ---------------------------------------------------------------------------
__global__ void k_scatter(const uint32_t* __restrict__ keysIn, const float* __restrict__ payIn,
                          uint32_t* __restrict__ keysOut, float* __restrict__ payOut,
                          const uint32_t* __restrict__ offsets, int shift, int numBlocks) {
    __shared__ uint16_t rank[256 * 256];  // [digit][thread]
    __shared__ uint32_t baseOff[256];
    const int tid = threadIdx.x;
    const int blk = blockIdx.x;

    baseOff[tid] = offsets[(size_t)tid * numBlocks + blk];
    for (int i = tid; i < 256 * 256; i += TPB) rank[i] = 0;
    __syncthreads();

    const long long base = (long long)blk * EPB + (long long)tid * IPT;
    uint32_t k[IPT];
    {
        const uint4* k4 = (const uint4*)(keysIn + base);
#pragma unroll
        for (int q = 0; q < 4; q++) {
            uint4 kk = k4[q];
            k[4 * q + 0] = kk.x; k[4 * q + 1] = kk.y; k[4 * q + 2] = kk.z; k[4 * q + 3] = kk.w;
        }
    }
#pragma unroll
    for (int j = 0; j < IPT; j++) {
        int d = (k[j] >> shift) & 255;
        rank[d * 256 + tid] = (uint16_t)(rank[d * 256 + tid] + 1);
    }
    __syncthreads();
    {   // thread `tid` owns digit row `tid`: serial exclusive scan over 256 thread columns
        const int row = tid * 256;
        uint32_t run = 0;
        for (int t = 0; t < 256; t++) {
            uint16_t c = rank[row + t];
            rank[row + t] = (uint16_t)run;
            run += c;
        }
    }
    __syncthreads();

    float p[IPT];
    {
        const float4* p4 = (const float4*)(payIn + base);
#pragma unroll
        for (int q = 0; q < 4; q++) {
            float4 pp = p4[q];
            p[4 * q + 0] = pp.x; p[4 * q + 1] = pp.y; p[4 * q + 2] = pp.z; p[4 * q + 3] = pp.w;
        }
    }
#pragma unroll
    for (int j = 0; j < IPT; j++) {
        int d = (k[j] >> shift) & 255;
        uint32_t r = rank[d * 256 + tid];
        rank[d * 256 + tid] = (uint16_t)(r + 1);
        uint32_t pos = baseOff[d] + r;
        keysOut[pos] = k[j];
        payOut[pos] = p[j];
    }
}

// ---------------------------------------------------------------------------
// Suffix-LSE stage A: per 4096-element block, (max, sum exp(x-max)) pair.
// ---------------------------------------------------------------------------
__global__ void k_block_pairs(const float* __restrict__ x, float2* __restrict__ pairs) {
    __shared__ float red[TPB];
    const int tid = threadIdx.x;
    const long long base = (long long)blockIdx.x * EPB + (long long)tid * IPT;
    const float4* x4 = (const float4*)(x + base);
    float v[IPT];
#pragma unroll
    for (int q = 0; q < 4; q++) {
        float4 t = x4[q];
        v[4 * q + 0] = t.x; v[4 * q + 1] = t.y; v[4 * q + 2] = t.z; v[4 * q + 3] = t.w;
    }
    float m = -INFINITY;
#pragma unroll
    for (int j = 0; j < IPT; j++) m = fmaxf(m, v[j]);
    red[tid] = m;
    __syncthreads();
    for (int off = TPB / 2; off > 0; off >>= 1) {
        if (tid < off) red[tid] = fmaxf(red[tid], red[tid + off]);
        __syncthreads();
    }
    float bm = red[0];
    __syncthreads();
    float s = 0.0f;
#pragma unroll
    for (int j = 0; j < IPT; j++) s += __expf(v[j] - bm);
    red[tid] = s;
    __syncthreads();
    for (int off = TPB / 2; off > 0; off >>= 1) {
        if (tid < off) red[tid] += red[tid + off];
        __syncthreads();
    }
    if (tid == 0) pairs[blockIdx.x] = make_float2(bm, red[0]);
}

// ---------------------------------------------------------------------------
// Suffix-LSE stage B: reverse-EXCLUSIVE scan of the block pairs (one block).
// tails[b] = combine of pairs[b+1 .. NB-1].
// ---------------------------------------------------------------------------
__global__ void k_block_tails(const float2* __restrict__ pairs, float2* __restrict__ tails,
                              int numBlocks) {
    __shared__ float am[TPB], asv[TPB];
    const int tid = threadIdx.x;
    const int ipc = numBlocks / TPB;  // == BPT
    float pm[BPT], ps[BPT];
    float cm = -INFINITY, cs = 0.0f;
#pragma unroll
    for (int j = 0; j < BPT; j++) {
        float2 pp = pairs[tid * ipc + j];
        pm[j] = pp.x; ps[j] = pp.y;
        lse_combine(cm, cs, pp.x, pp.y, cm, cs);
    }
    am[tid] = cm; asv[tid] = cs;
    __syncthreads();
    float im = cm, iv = cs;  // inclusive suffix over thread aggregates
    for (int off = 1; off < TPB; off <<= 1) {
        float om = -INFINITY, os = 0.0f;
        if (tid + off < TPB) { om = am[tid + off]; os = asv[tid + off]; }
        __syncthreads();
        lse_combine(im, iv, om, os, im, iv);
        am[tid] = im; asv[tid] = iv;
        __syncthreads();
    }
    float tm = -INFINITY, ts = 0.0f;
    if (tid + 1 < TPB) { tm = am[tid + 1]; ts = asv[tid + 1]; }
    float rm = tm, rs = ts;
    for (int j = BPT - 1; j >= 0; j--) {
        tails[tid * ipc + j] = make_float2(rm, rs);
        lse_combine(rm, rs, pm[j], ps[j], rm, rs);
    }
}

// ---------------------------------------------------------------------------
// Suffix-LSE stage C (WMMA): per block, 16-wide suffix sums of exp(x - M_seg)
// via V_WMMA_F32_16X16X4_F32 chains (A = upper-triangular ones, 4 K-panels;
// B = 16 independent segments), then exact online-softmax merge with the
// micro-segment/block tails and one v_log per element. Emits Σ lse per block.
// ---------------------------------------------------------------------------
__global__ void k_suffix_lse(const float* __restrict__ x, const float2* __restrict__ tails,
                             float* __restrict__ partialLse) {
    __shared__ float sExp[EPB];          // exp(x - M_seg), 16KB
    __shared__ float sM[TPB];            // micro-segment maxes
    __shared__ float sPm[TPB], sPs[TPB]; // pair suffix-scan arrays
    __shared__ float sTm[TPB], sTs[TPB]; // per-micro-segment tail pairs
    __shared__ float sRed[TPB];
    const int tid = threadIdx.x;
    const int blk = blockIdx.x;
    const long long base = (long long)blk * EPB + (long long)tid * IPT;
    const float4* x4 = (const float4*)(x + base);
    float v[IPT];
#pragma unroll
    for (int q = 0; q < 4; q++) {
        float4 t = x4[q];
        v[4 * q + 0] = t.x; v[4 * q + 1] = t.y; v[4 * q + 2] = t.z; v[4 * q + 3] = t.w;
    }
    float M = -INFINITY;
#pragma unroll
    for (int j = 0; j < IPT; j++) M = fmaxf(M, v[j]);
    float s = 0.0f;
#pragma unroll
    for (int j = 0; j < IPT; j++) {
        float e = __expf(v[j] - M);
        sExp[tid * IPT + j] = e;
        s += e;
    }
    sM[tid] = M; sPm[tid] = M; sPs[tid] = s;
    __syncthreads();

    // inclusive suffix scan over the 256 micro-segment pairs
    float im = M, iv = s;
    for (int off = 1; off < TPB; off <<= 1) {
        float om = -INFINITY, os = 0.0f;
        if (tid + off < TPB) { om = sPm[tid + off]; os = sPs[tid + off]; }
        __syncthreads();
        lse_combine(im, iv, om, os, im, iv);
        sPm[tid] = im; sPs[tid] = iv;
        __syncthreads();
    }
    // tail for micro-segment tid = (suffix of later segments) ⊕ (block tail)
    float2 T = tails[blk];
    float tm = -INFINITY, ts = 0.0f;
    if (tid + 1 < TPB) { tm = sPm[tid + 1]; ts = sPs[tid + 1]; }
    lse_combine(tm, ts, T.x, T.y, tm, ts);
    sTm[tid] = tm; sTs[tid] = ts;
    __syncthreads();

    // ---- WMMA suffix sums: each wave handles 32 segments = 2 x (16 segs) ----
    const int lane = tid & 31;
    const int wave = tid >> 5;
    const bool hi = lane >= 16;
    const int l16 = lane & 15;
    float local = 0.0f;
#pragma unroll
    for (int g = 0; g < 2; g++) {
        const int n = wave * 32 + g * 16 + l16;  // segment index for this lane's B column
        v8f acc = {0.f, 0.f, 0.f, 0.f, 0.f, 0.f, 0.f, 0.f};
#pragma unroll
        for (int p = 0; p < 4; p++) {
            const int ka = 4 * p + (hi ? 2 : 0);
            v2f bv;
            bv.x = sExp[n * IPT + ka];
            bv.y = sExp[n * IPT + ka + 1];
            v2f av;  // upper-triangular ones: A[m][k] = (k >= m)
            av.x = (ka >= l16) ? 1.0f : 0.0f;
            av.y = (ka + 1 >= l16) ? 1.0f : 0.0f;
            acc = __builtin_amdgcn_wmma_f32_16x16x4_f32(
                false, av, false, bv, (short)0, acc, false, false);
        }
        // lane holds suffix sums at positions r (+8 if hi) of segment n
        const float Mn = sM[n];
        const float ttm = sTm[n];
        const float tts = sTs[n];
        const float mstar = fmaxf(Mn, ttm);
        const float ea = __expf(Mn - mstar);
        const float eb = tts * __expf(ttm - mstar);
#pragma unroll
        for (int r = 0; r < 8; r++) {
            float S = acc[r];
            local += mstar + __logf(fmaxf(S * ea + eb, 1.0e-30f));
        }
    }
    sRed[tid] = local;
    __syncthreads();
    for (int off = TPB / 2; off > 0; off >>= 1) {
        if (tid < off) sRed[tid] += sRed[tid + off];
        __syncthreads();
    }
    if (tid == 0) partialLse[blk] = sRed[0];
}

// ---------------------------------------------------------------------------
// Final: deterministic reductions of the partials, loss, NaN override.
// ---------------------------------------------------------------------------
__global__ void k_final(const float* __restrict__ partialSum, const float* __restrict__ partialLse,
                        const int* __restrict__ nanFlag, float* __restrict__ out,
                        int numBlocks, int n) {
    __shared__ float red[TPB];
    const int tid = threadIdx.x;
    float a = 0.0f, b = 0.0f;
    for (int i = tid; i < numBlocks; i += TPB) {
        a += partialSum[i];
        b += partialLse[i];
    }
    red[tid] = a;
    __syncthreads();
    for (int off = TPB / 2; off > 0; off >>= 1) {
        if (tid < off) red[tid] += red[tid + off];
        __syncthreads();
    }
    float sumScores = red[0];
    __syncthreads();
    red[tid] = b;
    __syncthreads();
    for (int off = TPB / 2; off > 0; off >>= 1) {
        if (tid < off) red[tid] += red[tid + off];
        __syncthreads();
    }
    if (tid == 0) {
        float loss = (red[0] - sumScores) / (float)n;
        if (*nanFlag != 0) loss = 0.0f;
        out[0] = loss;
    }
}

// ---------------------------------------------------------------------------
extern "C" void kernel_launch(void* const* d_in, const int* in_sizes, int n_in,
                              void* d_out, int out_size, void* d_ws, size_t ws_size,
                              hipStream_t stream) {
    (void)n_in; (void)out_size; (void)ws_size;
    const float* scores = (const float*)d_in[0];
    const float* labels = (const float*)d_in[1];
    const int n = in_sizes[0];
    const int NB = n / EPB;  // 4096 for N = 2^24

    char* ws = (char*)d_ws;
    size_t off = 0;
    auto alloc = [&](size_t bytes) -> void* {
        void* p = (void*)(ws + off);
        off = (off + bytes + 255) & ~(size_t)255;
        return p;
    };
    uint32_t* keysA = (uint32_t*)alloc((size_t)n * 4);
    uint32_t* keysB = (uint32_t*)alloc((size_t)n * 4);
    float*    payA  = (float*)alloc((size_t)n * 4);
    float*    payB  = (float*)alloc((size_t)n * 4);
    uint32_t* counts   = (uint32_t*)alloc((size_t)256 * NB * 4);
    uint32_t* binTotal = (uint32_t*)alloc(256 * 4);
    uint32_t* binStart = (uint32_t*)alloc(256 * 4);
    float2*   pairs = (float2*)alloc((size_t)NB * 8);
    float2*   tails = (float2*)alloc((size_t)NB * 8);
    float* partialSum = (float*)alloc((size_t)NB * 4);
    float* partialLse = (float*)alloc((size_t)NB * 4);
    int*   nanFlag = (int*)alloc(4);

    hipMemsetAsync(nanFlag, 0, 4, stream);
    k_prep<<<NB, TPB, 0, stream>>>(labels, scores, keysA, payA, partialSum, nanFlag);

    uint32_t* ki = keysA; float* pi = payA;
    uint32_t* ko = keysB; float* po = payB;
    for (int pass = 0; pass < 4; pass++) {
        const int shift = pass * 8;
        k_hist<<<NB, TPB, 0, stream>>>(ki, counts, shift, NB);
        k_bin_total<<<256, TPB, 0, stream>>>(counts, binTotal, NB);
        k_bin_scan<<<1, 256, 0, stream>>>(binTotal, binStart);
        k_offsets<<<256, TPB, 0, stream>>>(counts, binStart, NB);
        k_scatter<<<NB, TPB, 0, stream>>>(ki, pi, ko, po, counts, shift, NB);
        uint32_t* tk = ki; ki = ko; ko = tk;
        float* tp = pi; pi = po; po = tp;
    }
    // after 4 passes the sorted (label-descending) scores are in `pi` (== payA)

    k_block_pairs<<<NB, TPB, 0, stream>>>(pi, pairs);
    k_block_tails<<<1, TPB, 0, stream>>>(pairs, tails, NB);
    k_suffix_lse<<<NB, TPB, 0, stream>>>(pi, tails, partialLse);
    k_final<<<1, TPB, 0, stream>>>(partialSum, partialLse, nanFlag, (float*)d_out, NB, n);
}